// ModelNew_50388556316976
// MI455X (gfx1250) — compile-verified
//
#include <hip/hip_runtime.h>
#include <hip/hip_bf16.h>

// ConvTranspose2d(64->64, k=3, s=1, p=0) on (2,64,1024,1024) fp32 -> (2,64,1026,1026)
// Implicit GEMM with V_WMMA_F32_16X16X4_F32 (exact fp32), LDS-staged input tiles,
// per-lane pre-packed weights (2x global_load_b128 per 4-channel K-step).
//
// out[n,o,oy,ox] = sum_{i,kh,kw} x[n,i,oy-kh,ox-kw] * w[i,o,kh,kw]   (x zero-padded)

#define H_IN   1024
#define W_IN   1024
#define H_OUT  1026
#define W_OUT  1026
#define CIN    64
#define COUT   64
#define NB     2

// LDS input tile: 64 channels x 10 rows x 18 cols of f32
#define TR 10   // rows (8 oy rows + 2 halo)
#define TC 18   // cols (16 ox + 2 halo)

typedef __attribute__((ext_vector_type(2))) float v2f;
typedef __attribute__((ext_vector_type(4))) float v4f;
typedef __attribute__((ext_vector_type(8))) float v8f;

// --- Kernel 1: pack weights per-lane ---
// wpack[((kk*16 + q)*32 + lane)*8 + i]  (kk=kh*3+kw, q=c4/4, i=2*t+j) holds
//   w[c = 4q + 2*(lane>>4) + j,  o = t*16 + (lane&15),  kh, kw]
// so in the main loop each lane fetches its 8 B-values for one K-quad as 2x b128.
__global__ __launch_bounds__(256) void convt_wpack(const float* __restrict__ w,
                                                   float* __restrict__ wp) {
    int idx = blockIdx.x * 256 + threadIdx.x;          // 9*16*32*8 = 36864 total
    if (idx >= 9 * CIN * COUT) return;
    int i    = idx & 7;
    int lane = (idx >> 3) & 31;
    int q    = (idx >> 8) & 15;
    int kk   = idx >> 12;                              // 0..8
    int kh   = kk / 3;
    int kw   = kk - kh * 3;
    int t    = i >> 1;
    int j    = i & 1;
    int c    = 4 * q + 2 * (lane >> 4) + j;
    int o    = t * 16 + (lane & 15);
    wp[idx] = w[((c * COUT + o) * 3 + kh) * 3 + kw];
}

// --- Kernel 2: main conv-transpose ---
// Grid: x = ox tiles (65 of 16), y = oy blocks (129 of 8), z = batch (2).
// Block: 256 threads = 8 wave32; wave w computes oy row (oy0+w), 16 ox, all 64 och.
__global__ __launch_bounds__(256) void convt_main(const float* __restrict__ x,
                                                  const float* __restrict__ wpack,
                                                  float* __restrict__ out) {
    __shared__ float xs[CIN * TR * TC];                // 11520 f32 = 45KB

    const int ox0 = blockIdx.x * 16;
    const int oy0 = blockIdx.y * 8;
    const int n   = blockIdx.z;
    const int tid = threadIdx.x;

    // ---- Stage input tile into LDS (zero-fill out-of-bounds halo) ----
    const float* xn = x + (size_t)n * CIN * H_IN * W_IN;
    #pragma unroll 1
    for (int idx = tid; idx < CIN * TR * TC; idx += 256) {
        int c   = idx / (TR * TC);
        int rem = idx - c * (TR * TC);
        int r   = rem / TC;
        int col = rem - r * TC;
        int iy  = oy0 - 2 + r;
        int ix  = ox0 - 2 + col;
        float v = 0.0f;
        if ((unsigned)iy < (unsigned)H_IN && (unsigned)ix < (unsigned)W_IN)
            v = xn[((size_t)c * H_IN + iy) * W_IN + ix];
        xs[idx] = v;
    }
    __syncthreads();

    const int lane  = tid & 31;
    const int wid   = tid >> 5;        // wave id = oy row within block
    const int oy    = oy0 + wid;
    const int lhalf = lane >> 4;       // 0: K=0,1  1: K=2,3   (f32 16x4 A layout)
    const int lm    = lane & 15;       // pixel index (A rows) / och-within-tile (B cols)

    v8f acc[4];
    #pragma unroll
    for (int t = 0; t < 4; ++t) acc[t] = (v8f){0.f,0.f,0.f,0.f,0.f,0.f,0.f,0.f};

    // ---- K-loop: (kh,kw) x channel quads; 144 steps x 4 WMMA = 576 WMMAs ----
    for (int kh = 0; kh < 3; ++kh) {
        const int r = wid + 2 - kh;                    // LDS row, in [0, 9]
        for (int kw = 0; kw < 3; ++kw) {
            const int col   = lm + 2 - kw;             // LDS col, in [0, 17]
            const int abase = (2 * lhalf) * (TR * TC) + r * TC + col;
            // per-lane packed weight pointer for this (kh,kw)
            const float* wl = wpack + ((size_t)((kh * 3 + kw) * 16) * 32 + lane) * 8;
            #pragma unroll 4
            for (int c4 = 0; c4 < CIN; c4 += 4) {
                // A: x[n, c4+2*lhalf + {0,1}, oy-kh, ox-kw] for 16 pixels
                v2f a;
                a.x = xs[abase + c4 * (TR * TC)];
                a.y = xs[abase + c4 * (TR * TC) + (TR * TC)];
                // B: 8 per-lane floats as two b128 loads (q = c4/4, stride 256 floats)
                const float* bp = wl + (c4 >> 2) * 256;
                v4f f4a = *(const v4f*)(bp);
                v4f f4b = *(const v4f*)(bp + 4);
                v2f b0, b1, b2, b3;
                b0.x = f4a.x; b0.y = f4a.y;
                b1.x = f4a.z; b1.y = f4a.w;
                b2.x = f4b.x; b2.y = f4b.y;
                b3.x = f4b.z; b3.y = f4b.w;
                acc[0] = __builtin_amdgcn_wmma_f32_16x16x4_f32(
                    false, a, false, b0, (short)0, acc[0], false, false);
                acc[1] = __builtin_amdgcn_wmma_f32_16x16x4_f32(
                    false, a, false, b1, (short)0, acc[1], false, false);
                acc[2] = __builtin_amdgcn_wmma_f32_16x16x4_f32(
                    false, a, false, b2, (short)0, acc[2], false, false);
                acc[3] = __builtin_amdgcn_wmma_f32_16x16x4_f32(
                    false, a, false, b3, (short)0, acc[3], false, false);
            }
        }
    }

    // ---- Transpose D through LDS so global stores are lane-contiguous ----
    __syncthreads();                    // everyone done reading xs; reuse it
    float* os = xs + wid * 1024;        // per-wave 16px x 64och region (8*1024 <= 11520)

    // D layout: acc[t] element e -> pixel = e + 8*lhalf, och = t*16 + lm
    #pragma unroll
    for (int t = 0; t < 4; ++t) {
        float* dst = os + (t * 16 + lm) * 16 + 8 * lhalf;   // [och][px], 32B aligned
        v4f lo, hi;
        lo.x = acc[t][0]; lo.y = acc[t][1]; lo.z = acc[t][2]; lo.w = acc[t][3];
        hi.x = acc[t][4]; hi.y = acc[t][5]; hi.z = acc[t][6]; hi.w = acc[t][7];
        *(v4f*)(dst)     = lo;
        *(v4f*)(dst + 4) = hi;
    }
    // no barrier needed: each wave reads back only its own region

    if (oy < H_OUT) {
        const int och_lo = lane >> 3;        // 0..3
        const int px     = (lane & 7) * 2;   // 0,2,...,14
        const int ox     = ox0 + px;         // even; W_OUT even -> one guard covers both
        if (ox < W_OUT) {
            #pragma unroll 4
            for (int g = 0; g < 16; ++g) {
                int och = g * 4 + och_lo;
                v2f v = *(const v2f*)(os + och * 16 + px);
                size_t base = ((size_t)(n * COUT + och) * H_OUT + oy) * W_OUT + ox;
                // streaming output: non-temporal so it doesn't evict weights/x halo in L2
                __builtin_nontemporal_store(v, (v2f*)(out + base));
            }
        }
    }
}

extern "C" void kernel_launch(void* const* d_in, const int* in_sizes, int n_in,
                              void* d_out, int out_size, void* d_ws, size_t ws_size,
                              hipStream_t stream) {
    const float* x = (const float*)d_in[0];   // (2,64,1024,1024) fp32
    const float* w = (const float*)d_in[1];   // (64,64,3,3) fp32
    float* out = (float*)d_out;               // (2,64,1026,1026) fp32
    float* wp  = (float*)d_ws;                // 9*64*64 f32 = 147456 B scratch

    convt_wpack<<<(9 * CIN * COUT + 255) / 256, 256, 0, stream>>>(w, wp);

    dim3 grid((W_OUT + 15) / 16,   // 65 ox tiles
              (H_OUT + 7) / 8,     // 129 oy blocks
              NB);                 // 2 batches
    convt_main<<<grid, 256, 0, stream>>>(x, wp, out);
}